// LinearAttn_82721070121774
// MI455X (gfx1250) — compile-verified
//
#include <hip/hip_runtime.h>

// Problem constants (match the reference)
#define N_B    8
#define T_LEN  2048
#define TQ_LEN 2048
#define D_DIM  1024
#define H_DIM  1024
#define P_LEN  16
#define S_LEN  (P_LEN + T_LEN)   // 2064
#define S_CHUNKS 8               // 2064 / 8 = 258 rows per chunk
#define S_PER_CHUNK (S_LEN / S_CHUNKS)
#define M_PAD  16                // WMMA M dimension (batch 8 zero-padded to 16)

typedef __attribute__((ext_vector_type(2))) float v2f;
typedef __attribute__((ext_vector_type(8))) float v8f;

__device__ __forceinline__ float wave_reduce_f32(float v) {
#pragma unroll
    for (int off = 16; off > 0; off >>= 1)
        v += __shfl_xor(v, off, 32);
    return v;
}

// -------- Kernel 1: vlast[n,s] = seq[n,s,:] . Wv[D-1,:] --------------------
// One wave32 per (n,s) row; Wv last row staged in LDS; float4 (b128) loads.
__global__ void vlast_kernel(const float* __restrict__ input_seq,
                             const float* __restrict__ prompts,
                             const float* __restrict__ Wv,
                             const int*   __restrict__ task_id,
                             float*       __restrict__ vlast) {
    __shared__ float4 wlds[D_DIM / 4];
    const int tid = threadIdx.x;                      // blockDim.x == 256
    const float4* wrow = (const float4*)(Wv + (size_t)(D_DIM - 1) * D_DIM);
    wlds[tid] = wrow[tid];
    __syncthreads();

    const int wave = tid >> 5, lane = tid & 31;
    const int row  = blockIdx.x * 8 + wave;           // 0 .. N_B*S_LEN-1
    if (row >= N_B * S_LEN) return;
    const int n = row / S_LEN, s = row % S_LEN;

    const float* src = (s < P_LEN)
        ? prompts   + ((size_t)task_id[0] * P_LEN + s)          * D_DIM
        : input_seq + ((size_t)n * T_LEN + (size_t)(s - P_LEN)) * D_DIM;
    const float4* rp = (const float4*)src;

    float acc = 0.0f;
#pragma unroll
    for (int i = lane; i < D_DIM / 4; i += 32) {      // 8 x b128 per lane
        float4 x = rp[i];
        float4 w = wlds[i];
        acc += x.x * w.x + x.y * w.y + x.z * w.z + x.w * w.w;
    }
    acc = wave_reduce_f32(acc);
    if (lane == 0) vlast[row] = acc;
}

// -------- Kernel 2: u_part[c,n,d] = sum_{s in chunk c} seq[n,s,d]*vlast[n,s]
// Deterministic (no float atomics): partials per s-chunk, reduced afterwards.
// seq re-read here hits L2 (67 MB fits in the 192 MB L2).
__global__ void u_part_kernel(const float* __restrict__ input_seq,
                              const float* __restrict__ prompts,
                              const float* __restrict__ vlast,
                              const int*   __restrict__ task_id,
                              float*       __restrict__ u_part) {
    const int n  = blockIdx.z;
    const int d  = blockIdx.x * 256 + threadIdx.x;    // coalesced over d
    const int c  = blockIdx.y;
    const int s0 = c * S_PER_CHUNK;
    const int ptask = task_id[0];

    float acc = 0.0f;
    for (int s = s0; s < s0 + S_PER_CHUNK; ++s) {
        const float* src = (s < P_LEN)
            ? prompts   + ((size_t)ptask * P_LEN + s)               * D_DIM
            : input_seq + ((size_t)n * T_LEN + (size_t)(s - P_LEN)) * D_DIM;
        acc += src[d] * vlast[n * S_LEN + s];
    }
    u_part[((size_t)c * N_B + n) * D_DIM + d] = acc;
}

// Reduce the 8 partials into a zero-padded [16,1024] A-matrix for WMMA
// (rows 8..15 written as exact zeros -> no exec masking in the WMMA loop).
__global__ void u_reduce_pad_kernel(const float* __restrict__ u_part,
                                    float*       __restrict__ upad) {
    const int i = blockIdx.x * 256 + threadIdx.x;     // 0 .. M_PAD*D_DIM-1
    float acc = 0.0f;
    if (i < N_B * D_DIM) {
#pragma unroll
        for (int c = 0; c < S_CHUNKS; ++c)
            acc += u_part[(size_t)c * (N_B * D_DIM) + i];
    }
    upad[i] = acc;                                    // rows 8..15 -> 0
}

// -------- Kernels 3/4: D(16xN) = Apad(16xK) @ B(KxN) via V_WMMA_F32_16X16X4_F32
// Apad is [16,1024] with rows 8..15 zero. BT=true: B[k,n] = Bm[n*1024+k]
// (B = Wk^T). BT=false: B[k,n] = Bm[k*1024+n] (B = Wq as stored).
// STORE_ALL=true writes all 16 rows (padding rows are exact zeros, keeping the
// next stage's A operand padded); STORE_ALL=false writes only the 8 real rows.
// Layouts per CDNA5 ISA 7.12.2:
//   A 16x4 : lane l -> M=l%16, VGPR j -> K = 2*(l/16)+j
//   B 4x16 : lane l -> N=l%16, VGPR j -> K = 2*(l/16)+j
//   C/D    : lane l -> N=l%16, VGPR v -> M = v + 8*(l/16)
template <bool BT, bool STORE_ALL>
__global__ void small_gemm_wmma(const float* __restrict__ A,   // [16,1024] padded
                                const float* __restrict__ Bm,  // [1024,1024]
                                float*       __restrict__ Dout) {
    const int lane = threadIdx.x;        // blockDim.x == 32, EXEC all-ones
    const int half = lane >> 4;          // 0 or 1
    const int lm   = lane & 15;
    const int n0   = blockIdx.x * 16;

    const float* arow = A + (size_t)lm * 1024;        // padded: unconditional

    v8f c = {};
    for (int k0 = 0; k0 < 1024; k0 += 4) {
        const int ka = k0 + 2 * half;
        v2f a;
        a[0] = arow[ka];                              // one b64 load
        a[1] = arow[ka + 1];
        v2f b;
        if (BT) {
            const float* bp = Bm + (size_t)(n0 + lm) * 1024 + ka;
            b[0] = bp[0];                             // one b64 load
            b[1] = bp[1];
        } else {
            b[0] = Bm[(size_t)ka * 1024 + n0 + lm];
            b[1] = Bm[(size_t)(ka + 1) * 1024 + n0 + lm];
        }
        // 8 args: (neg_a, A, neg_b, B, c_mod, C, reuse_a, reuse_b)
        c = __builtin_amdgcn_wmma_f32_16x16x4_f32(false, a, false, b,
                                                  (short)0, c, false, false);
    }
    if (STORE_ALL) {
#pragma unroll
        for (int v = 0; v < 8; ++v)                   // rows 0..15
            Dout[(size_t)(v + 8 * half) * 1024 + n0 + lm] = c[v];
    } else if (half == 0) {
#pragma unroll
        for (int v = 0; v < 8; ++v)                   // rows 0..7 only
            Dout[(size_t)v * 1024 + n0 + lm] = c[v];
    }
}

// -------- Kernel 6: out[n,q] = scale * cross[n,q,:] . r[n,:] ---------------
__global__ void out_kernel(const float* __restrict__ cross,
                           const float* __restrict__ r,
                           float*       __restrict__ out) {
    __shared__ float4 rlds[D_DIM / 4];
    const int tid  = threadIdx.x;                     // blockDim.x == 256
    const int wave = tid >> 5, lane = tid & 31;
    const int row  = blockIdx.x * 8 + wave;           // 0 .. N_B*TQ_LEN-1
    const int n    = row / TQ_LEN;                    // TQ%8==0: same n per block

    const float4* rr = (const float4*)(r + (size_t)n * D_DIM);
    rlds[tid] = rr[tid];
    __syncthreads();

    const float4* cp = (const float4*)(cross + (size_t)row * D_DIM);
    float acc = 0.0f;
#pragma unroll
    for (int i = lane; i < D_DIM / 4; i += 32) {
        float4 x = cp[i];
        float4 w = rlds[i];
        acc += x.x * w.x + x.y * w.y + x.z * w.z + x.w * w.w;
    }
    acc = wave_reduce_f32(acc);
    if (lane == 0) out[row] = acc * (1.0f / 32.0f);   // scale = 1/sqrt(1024)
}

extern "C" void kernel_launch(void* const* d_in, const int* in_sizes, int n_in,
                              void* d_out, int out_size, void* d_ws, size_t ws_size,
                              hipStream_t stream) {
    const float* input_seq = (const float*)d_in[0];   // [8,2048,1024]
    const float* cross     = (const float*)d_in[1];   // [8,2048,1024]
    const float* prompts   = (const float*)d_in[2];   // [1,16,1024]
    const float* Wq        = (const float*)d_in[3];   // [1024,1024]
    const float* Wk        = (const float*)d_in[4];   // [1024,1024]
    const float* Wv        = (const float*)d_in[5];   // [1024,1024]
    const int*   task_id   = (const int*)d_in[6];     // scalar
    float* out = (float*)d_out;                       // [8,2048]

    // Workspace carve-up (all regions fully overwritten every call).
    float* vlast  = (float*)d_ws;                     // N*S            = 16512
    float* u_part = vlast  + N_B * S_LEN;             // 8*N*D          = 65536
    float* upad   = u_part + S_CHUNKS * N_B * D_DIM;  // 16*D           = 16384
    float* kwpad  = upad   + M_PAD * D_DIM;           // 16*H           = 16384
    float* r      = kwpad  + M_PAD * H_DIM;           // N*D            = 8192

    // 1) vlast = seq . Wv[last row]          (HBM-bound: 67 MB)
    vlast_kernel<<<(N_B * S_LEN) / 8, 256, 0, stream>>>(
        input_seq, prompts, Wv, task_id, vlast);

    // 2) u[n,d] = sum_s seq[n,s,d]*vlast[n,s]  (L2-bound re-read, deterministic)
    u_part_kernel<<<dim3(D_DIM / 256, S_CHUNKS, N_B), 256, 0, stream>>>(
        input_seq, prompts, vlast, task_id, u_part);
    u_reduce_pad_kernel<<<(M_PAD * D_DIM) / 256, 256, 0, stream>>>(u_part, upad);

    // 3) kwpad = Upad @ Wk^T   (WMMA f32 16x16x4, 64 tiles, padded output)
    small_gemm_wmma<true, true><<<H_DIM / 16, 32, 0, stream>>>(upad, Wk, kwpad);

    // 4) r = KWpad @ Wq        (WMMA f32 16x16x4, 64 tiles, real rows only)
    small_gemm_wmma<false, false><<<D_DIM / 16, 32, 0, stream>>>(kwpad, Wq, r);

    // 5) out = scale * cross . r             (HBM-bound: 67 MB)
    out_kernel<<<(N_B * TQ_LEN) / 8, 256, 0, stream>>>(cross, r, out);
}